// HadamardTransform_36240934044416
// MI455X (gfx1250) — compile-verified
//
#include <hip/hip_runtime.h>

// Radix-16 fast Walsh-Hadamard transform for N=4096 using
// V_WMMA_F32_16X16X4_F32 (CDNA5). H_4096 = (1/64) * H16 (x) H16 (x) H16,
// H16[i][j] = (-1)^popcount(i&j). Three batched 16x16 GEMM stages per row.
//
// Memory-bound design: each 4096-float row is read once from HBM, transformed
// in LDS/registers by one wave, written once (268 MB total -> ~11.5us floor
// at 23.3 TB/s).

typedef float v2f __attribute__((ext_vector_type(2)));
typedef float v8f __attribute__((ext_vector_type(8)));

#define HN 4096
#define ROWS_PER_BLOCK 4

__device__ __forceinline__ float hsgn(int i, int j) {
    return (__builtin_popcount(i & j) & 1) ? -1.0f : 1.0f;
}

__global__ __launch_bounds__(32 * ROWS_PER_BLOCK)
void fwht4096_wmma(const float* __restrict__ X, float* __restrict__ Y) {
    __shared__ float ldsAll[ROWS_PER_BLOCK][HN];

    const int tid  = threadIdx.x;
    const int lane = tid & 31;
    const int wv   = tid >> 5;
    const int n    = lane & 15;   // M (lanes 0-15) or M-16 (lanes 16-31); also N for B/C/D
    const int hi   = lane >> 4;   // 0 = low half-wave, 1 = high half-wave
    const int khi  = hi * 2;      // K offset within a K=4 step (ISA A/B layout)
    const int hi8  = hi * 8;      // C/D row offset for high half-wave

    const long row = (long)blockIdx.x * ROWS_PER_BLOCK + wv;
    const float* __restrict__ xr = X + row * HN;
    float*       __restrict__ yr = Y + row * HN;
    float* sm = ldsAll[wv];

    // Precompute the four K=4 fragments of H16 (identical layout whether H16 is
    // used as the A matrix (M=lane&15) or the B matrix (N=lane&15): symmetric).
    v2f hf[4];
#pragma unroll
    for (int kk = 0; kk < 4; ++kk) {
        hf[kk].x = hsgn(n, kk * 4 + khi);
        hf[kk].y = hsgn(n, kk * 4 + khi + 1);
    }

    // ---- Stage 1: transform along c (stride 1).  y1[ab][k] = sum_c x[ab][c]*H[c][k]
    // A = data tile (rows = 16 consecutive (a,b) pairs, K = c), B = H16.
#pragma unroll 4
    for (int t = 0; t < 16; ++t) {
        v8f d = {0.f, 0.f, 0.f, 0.f, 0.f, 0.f, 0.f, 0.f};
#pragma unroll
        for (int kk = 0; kk < 4; ++kk) {
            const int k0 = kk * 4;
            v2f a;
            a.x = xr[(t * 16 + n) * 16 + k0 + khi];
            a.y = xr[(t * 16 + n) * 16 + k0 + khi + 1];
            d = __builtin_amdgcn_wmma_f32_16x16x4_f32(
                    false, a, false, hf[kk], (short)0, d, false, false);
        }
#pragma unroll
        for (int r = 0; r < 8; ++r)
            sm[(t * 16 + r + hi8) * 16 + n] = d[r];
    }

    // ---- Stage 2: transform along b (stride 16), per a-slice, in place.
    // y2[a][i][k] = sum_b H[i][b] * y1[a][b][k]   =>  A = H16, B = slice.
#pragma unroll 2
    for (int aa = 0; aa < 16; ++aa) {
        v2f bf[4];
#pragma unroll
        for (int kk = 0; kk < 4; ++kk) {
            const int k0 = kk * 4;
            bf[kk].x = sm[aa * 256 + (k0 + khi) * 16 + n];
            bf[kk].y = sm[aa * 256 + (k0 + khi + 1) * 16 + n];
        }
        v8f d = {0.f, 0.f, 0.f, 0.f, 0.f, 0.f, 0.f, 0.f};
#pragma unroll
        for (int kk = 0; kk < 4; ++kk)
            d = __builtin_amdgcn_wmma_f32_16x16x4_f32(
                    false, hf[kk], false, bf[kk], (short)0, d, false, false);
#pragma unroll
        for (int r = 0; r < 8; ++r)
            sm[aa * 256 + (r + hi8) * 16 + n] = d[r];
    }

    // ---- Stage 3: transform along a (stride 256), fused 1/64 scale + store.
    // y[i][b][k] = (1/64) * sum_a H[i][a] * y2[a][b][k]
#pragma unroll 2
    for (int bb = 0; bb < 16; ++bb) {
        v2f bf[4];
#pragma unroll
        for (int kk = 0; kk < 4; ++kk) {
            const int k0 = kk * 4;
            bf[kk].x = sm[(k0 + khi) * 256 + bb * 16 + n];
            bf[kk].y = sm[(k0 + khi + 1) * 256 + bb * 16 + n];
        }
        v8f d = {0.f, 0.f, 0.f, 0.f, 0.f, 0.f, 0.f, 0.f};
#pragma unroll
        for (int kk = 0; kk < 4; ++kk)
            d = __builtin_amdgcn_wmma_f32_16x16x4_f32(
                    false, hf[kk], false, bf[kk], (short)0, d, false, false);
#pragma unroll
        for (int r = 0; r < 8; ++r)
            yr[(r + hi8) * 256 + bb * 16 + n] = d[r] * 0.015625f;  // * 2^-6
    }
}

extern "C" void kernel_launch(void* const* d_in, const int* in_sizes, int n_in,
                              void* d_out, int out_size, void* d_ws, size_t ws_size,
                              hipStream_t stream) {
    (void)n_in; (void)out_size; (void)d_ws; (void)ws_size;
    const float* X = (const float*)d_in[0];
    // d_in[1] (H) is unused: its entries are reproduced exactly by the
    // popcount sign generator + 2^-6 normalization.
    float* Y = (float*)d_out;

    const int rows   = in_sizes[0] / HN;          // 8192
    const int blocks = rows / ROWS_PER_BLOCK;     // 2048 blocks of 128 threads
    fwht4096_wmma<<<blocks, 32 * ROWS_PER_BLOCK, 0, stream>>>(X, Y);
}